// Decoder_71665824301075
// MI455X (gfx1250) — compile-verified
//
#include <hip/hip_runtime.h>
#include <hip/hip_bf16.h>

// CDNA5 / gfx1250. Wave32. Matrix ops via V_WMMA_F32_16X16X4_F32 (f32 matrix pipe,
// full precision — the model is f32 end to end).
//
//   Kernel 1: XW[b,t,:] = X[b,t,:] @ aW1_x^T + ab1   (time-invariant, hoisted out of scan)
//   Kernel 2: 16 workgroups (16-batch tiles), 512 threads (16 waves, 4/SIMD), 63-step scan:
//     u_d = d@Wd^T (waves 0-7) ; u_s = s@Ws^T (waves 8-15)    [WMMA, parallel]
//     z[b,t] = sum_k tanh(XW + u_d + u_s)*aW2[k]              [trans pass, L2 stream]
//     softmax over t (ab2 drops out), context = betas@X
//     gates = yup*Wih + d@Whh^T + b                           [WMMA, 2 col-tiles/wave,
//                                                              A-frags hoisted to regs]
//     LSTM elementwise update of d,s in LDS
//   Final: out[b] = [d,context] @ Wo^T + bo

typedef __attribute__((ext_vector_type(2))) float v2f;
typedef __attribute__((ext_vector_type(4))) float f4v;
typedef __attribute__((ext_vector_type(8))) float v8f;

#define TM1 63
#define HH  128
#define BT  16   // batch tile per workgroup (WMMA M)

__device__ __forceinline__ float fast_tanh(float x) {
#if __has_builtin(__builtin_amdgcn_tanhf)
    return __builtin_amdgcn_tanhf(x);
#elif __has_builtin(__builtin_amdgcn_tanh_f32)
    return __builtin_amdgcn_tanh_f32(x);
#else
    float ax = fabsf(x);
    float e  = __expf(-2.0f * ax);
    float t  = (1.0f - e) / (1.0f + e);
    return copysignf(t, x);
#endif
}

__device__ __forceinline__ float sigmoidf_(float x) {
    return 1.0f / (1.0f + __expf(-x));
}

// -------------------------------------------------------------------------
// Kernel 1: XW = X @ aW1_x^T + ab1, rows = B*TM1 = 16128, cols = 128.
// 512 threads = 16 waves: wave = {rowHalf 0/1} x {col tile 0..7}.
// -------------------------------------------------------------------------
__global__ __launch_bounds__(512) void xw_precompute(
    const float* __restrict__ X,    // [16128,128]
    const float* __restrict__ aW1,  // [128,384]
    const float* __restrict__ ab1,  // [128]
    float* __restrict__ XW)         // [16128,128]
{
    const int lane = threadIdx.x & 31;
    const int wave = threadIdx.x >> 5;     // 0..15
    const int l16  = lane & 15;
    const int half = lane >> 4;
    const int rowbase = blockIdx.x * 32 + (wave >> 3) * 16;
    const int obase   = (wave & 7) * 16;

    const float* Arow = X   + (size_t)(rowbase + l16) * HH;
    const float* Brow = aW1 + (size_t)(obase + l16) * 384 + 256;  // X-block of aW1

    v8f acc = {};
    for (int k = 0; k < HH; k += 4) {
        v2f a = *(const v2f*)(Arow + k + 2 * half);
        v2f b = *(const v2f*)(Brow + k + 2 * half);
        acc = __builtin_amdgcn_wmma_f32_16x16x4_f32(false, a, false, b,
                                                    (short)0, acc, false, false);
    }
    const float bias = ab1[obase + l16];
#pragma unroll
    for (int j = 0; j < 8; ++j) {
        int m = j + 8 * half;
        XW[(size_t)(rowbase + m) * HH + obase + l16] = acc[j] + bias;
    }
}

// -------------------------------------------------------------------------
// Kernel 2: full 63-step scan. 16 blocks x 512 threads. LDS-resident state.
// -------------------------------------------------------------------------
__global__ __launch_bounds__(512) void scan_kernel(
    const float* __restrict__ XW,   // [B*63,128] precomputed (incl. ab1)
    const float* __restrict__ X,    // [B,63,128]
    const float* __restrict__ Y,    // [B,63]
    const float* __restrict__ aW1,  // [128,384] (d-block @0, s-block @128)
    const float* __restrict__ aW2,  // [128]
    const float* __restrict__ Wih,  // [512]
    const float* __restrict__ Whh,  // [512,128]
    const float* __restrict__ bih,  // [512]
    const float* __restrict__ bhh,  // [512]
    const float* __restrict__ Wfc,  // [129]
    const float* __restrict__ bfc,  // [1]
    const float* __restrict__ Wo,   // [256]
    const float* __restrict__ bo,   // [1]
    float* __restrict__ out)        // [B]
{
    // padded strides (132, 528) to break 64-bank alignment on row-indexed loads
    __shared__ float dld [BT * 132];
    __shared__ float sld [BT * 132];
    __shared__ float uld [BT * 132];   // d-part of u
    __shared__ float uld2[BT * 132];   // s-part of u
    __shared__ float ctx [BT * 132];
    __shared__ float zld [BT * 64];
    __shared__ float gld [BT * 528];
    __shared__ float red [BT * 32];
    __shared__ float yup [BT];
    __shared__ float aW2L[128], WihL[512], bsumL[512], WfcL[132], WoL[256];

    const int tid  = threadIdx.x;          // 0..511
    const int lane = tid & 31;
    const int wave = tid >> 5;             // 0..15
    const int l16  = lane & 15;
    const int half = lane >> 4;
    const int bbase = blockIdx.x * BT;
    const int bb = tid >> 5;               // 0..15 (batch row, elementwise phases)
    const int jj = tid & 31;               // 0..31 (32 threads cooperate per row)

    for (int i = tid; i < 512; i += 512) { WihL[i] = Wih[i]; bsumL[i] = bih[i] + bhh[i]; }
    for (int i = tid; i < 128; i += 512) aW2L[i] = aW2[i];
    for (int i = tid; i < 129; i += 512) WfcL[i] = Wfc[i];
    for (int i = tid; i < 256; i += 512) WoL[i]  = Wo[i];
    for (int i = tid; i < BT * 132; i += 512) { dld[i] = 0.f; sld[i] = 0.f; ctx[i] = 0.f; }
    __syncthreads();

    for (int step = 0; step < TM1; ++step) {
        // -- phase 1: u_d = d@Wd^T (waves 0-7), u_s = s@Ws^T (waves 8-15) ----
        {
            const int   obase = (wave & 7) * 16;
            const bool  isS   = (wave >= 8);
            const float* Arow = (isS ? sld : dld) + l16 * 132;
            const float* Brow = aW1 + (size_t)(obase + l16) * 384 + (isS ? 128 : 0);
            float*       uout = isS ? uld2 : uld;
            v8f acc = {};
            for (int k = 0; k < HH; k += 4) {
                v2f a = *(const v2f*)(Arow + k + 2 * half);
                v2f b = *(const v2f*)(Brow + k + 2 * half);
                acc = __builtin_amdgcn_wmma_f32_16x16x4_f32(false, a, false, b,
                                                            (short)0, acc, false, false);
            }
#pragma unroll
            for (int j = 0; j < 8; ++j)
                uout[(j + 8 * half) * 132 + obase + l16] = acc[j];
        }
        __syncthreads();

        // -- phase 2: z[b][t] = sum_k tanh(XW + u_d + u_s)*aW2 ---------------
        for (int p = 0; p < 2; ++p) {
            int idx = tid + p * 512;
            if (idx < BT * TM1) {
                int b = idx / TM1, t = idx % TM1;
                const float* xw  = XW + ((size_t)(bbase + b) * TM1 + t) * HH;
                const float* urd = uld  + b * 132;
                const float* urs = uld2 + b * 132;
                float acc = 0.f;
                for (int k = 0; k < HH; k += 4) {
                    f4v x = *(const f4v*)(xw + k);
                    acc += fast_tanh(x.x + urd[k + 0] + urs[k + 0]) * aW2L[k + 0];
                    acc += fast_tanh(x.y + urd[k + 1] + urs[k + 1]) * aW2L[k + 1];
                    acc += fast_tanh(x.z + urd[k + 2] + urs[k + 2]) * aW2L[k + 2];
                    acc += fast_tanh(x.w + urd[k + 3] + urs[k + 3]) * aW2L[k + 3];
                }
                zld[b * 64 + t] = acc;
            }
        }
        __syncthreads();

        // -- phase 3: softmax over t (32 threads per batch row) --------------
        {
            float m = -1e30f;
            for (int t = jj; t < TM1; t += 32) m = fmaxf(m, zld[bb * 64 + t]);
            red[bb * 32 + jj] = m;
            __syncthreads();
            if (jj == 0) {
                float mm = red[bb * 32];
                for (int q = 1; q < 32; ++q) mm = fmaxf(mm, red[bb * 32 + q]);
                red[bb * 32] = mm;
            }
            __syncthreads();
            float mm = red[bb * 32];
            float ls = 0.f;
            for (int t = jj; t < TM1; t += 32) {
                float e = __expf(zld[bb * 64 + t] - mm);
                zld[bb * 64 + t] = e;
                ls += e;
            }
            __syncthreads();
            red[bb * 32 + jj] = ls;
            __syncthreads();
            if (jj == 0) {
                float s_ = 0.f;
                for (int q = 0; q < 32; ++q) s_ += red[bb * 32 + q];
                red[bb * 32] = 1.0f / s_;
            }
            __syncthreads();
            float inv = red[bb * 32];
            for (int t = jj; t < TM1; t += 32) zld[bb * 64 + t] *= inv;
        }
        __syncthreads();

        // -- phase 4: context = betas @ X  (thread: batch bb, 4 cols) --------
        {
            const float* Xr = X + (size_t)(bbase + bb) * TM1 * HH + jj * 4;
            f4v a0 = {};
            for (int t = 0; t < TM1; ++t) {
                float bet = zld[bb * 64 + t];
                a0 += bet * *(const f4v*)(Xr + t * HH);
            }
            *(f4v*)(ctx + bb * 132 + jj * 4) = a0;
        }
        __syncthreads();

        // -- phase 5: y_update[b] = Wfc[0]*y_t + Wfc[1:]·context + bfc -------
        {
            const float* cr = ctx + bb * 132 + jj * 4;
            float ps = 0.f;
#pragma unroll
            for (int q = 0; q < 4; ++q) ps += WfcL[1 + jj * 4 + q] * cr[q];
            red[bb * 32 + jj] = ps;
            __syncthreads();
            if (jj == 0) {
                float s_ = 0.f;
                for (int q = 0; q < 32; ++q) s_ += red[bb * 32 + q];
                float yt = Y[(size_t)(bbase + bb) * TM1 + step];
                yup[bb] = s_ + WfcL[0] * yt + bfc[0];
            }
        }
        __syncthreads();

        // -- phase 6: gates = yup*Wih + d@Whh^T + (bih+bhh) ------------------
        // A fragments (d-tile) hoisted into registers once, reused for 2 col-tiles.
        {
            v2f afrag[32];
            const float* Ad = dld + l16 * 132;
#pragma unroll
            for (int kk = 0; kk < 32; ++kk)
                afrag[kk] = *(const v2f*)(Ad + kk * 4 + 2 * half);
#pragma unroll
            for (int c = 0; c < 2; ++c) {
                int gbase = (wave + 16 * c) * 16;
                const float* Bw = Whh + (size_t)(gbase + l16) * HH;
                v8f acc = {};
#pragma unroll 8
                for (int kk = 0; kk < 32; ++kk) {
                    v2f b = *(const v2f*)(Bw + kk * 4 + 2 * half);
                    acc = __builtin_amdgcn_wmma_f32_16x16x4_f32(false, afrag[kk], false, b,
                                                                (short)0, acc, false, false);
                }
                int g = gbase + l16;
                float wih = WihL[g], bs = bsumL[g];
#pragma unroll
                for (int j = 0; j < 8; ++j) {
                    int m = j + 8 * half;
                    gld[m * 528 + g] = acc[j] + yup[m] * wih + bs;
                }
            }
        }
        __syncthreads();

        // -- phase 7: LSTM elementwise (gate order i,f,g,o) ------------------
        {
            int h0 = jj * 4;
#pragma unroll
            for (int q = 0; q < 4; ++q) {
                int h = h0 + q;
                float ig = sigmoidf_(gld[bb * 528 + h]);
                float fg = sigmoidf_(gld[bb * 528 + 128 + h]);
                float gg = fast_tanh (gld[bb * 528 + 256 + h]);
                float og = sigmoidf_(gld[bb * 528 + 384 + h]);
                float c_ = fg * sld[bb * 132 + h] + ig * gg;
                sld[bb * 132 + h] = c_;
                dld[bb * 132 + h] = og * fast_tanh(c_);
            }
        }
        __syncthreads();
    }

    // -- final: out[b] = Wo[:128]·d + Wo[128:]·context + bo -----------------
    {
        float ps = 0.f;
#pragma unroll
        for (int q = 0; q < 4; ++q) {
            int k = jj * 4 + q;
            ps += WoL[k] * dld[bb * 132 + k] + WoL[128 + k] * ctx[bb * 132 + k];
        }
        red[bb * 32 + jj] = ps;
        __syncthreads();
        if (jj == 0) {
            float s_ = 0.f;
            for (int q = 0; q < 32; ++q) s_ += red[bb * 32 + q];
            out[bbase + bb] = s_ + bo[0];
        }
    }
}

extern "C" void kernel_launch(void* const* d_in, const int* in_sizes, int n_in,
                              void* d_out, int out_size, void* d_ws, size_t ws_size,
                              hipStream_t stream) {
    const float* X   = (const float*)d_in[0];   // [256,63,128]
    const float* Y   = (const float*)d_in[1];   // [256,63]
    const float* aW1 = (const float*)d_in[2];   // [128,384]
    const float* ab1 = (const float*)d_in[3];   // [128]
    const float* aW2 = (const float*)d_in[4];   // [128]
    // d_in[5] = ab2: shifts softmax logits by a constant -> mathematically a no-op
    const float* Wih = (const float*)d_in[6];   // [512,1]
    const float* Whh = (const float*)d_in[7];   // [512,128]
    const float* bih = (const float*)d_in[8];   // [512]
    const float* bhh = (const float*)d_in[9];   // [512]
    const float* Wfc = (const float*)d_in[10];  // [1,129]
    const float* bfc = (const float*)d_in[11];  // [1]
    const float* Wo  = (const float*)d_in[12];  // [1,256]
    const float* bo  = (const float*)d_in[13];  // [1]
    float* out = (float*)d_out;                 // [256]

    float* XW = (float*)d_ws;                   // [256*63,128] = 8.25 MB

    // Kernel 1: 16128 rows / 32 = 504 blocks; 16 waves = 2 row-tiles x 8 col-tiles
    xw_precompute<<<dim3(504), dim3(512), 0, stream>>>(X, aW1, ab1, XW);

    // Kernel 2: 256 batches / 16 = 16 workgroups, each runs the full scan
    scan_kernel<<<dim3(16), dim3(512), 0, stream>>>(
        XW, X, Y, aW1, aW2, Wih, Whh, bih, bhh, Wfc, bfc, Wo, bo, out);
}